// ODEFunc_77318001262682
// MI455X (gfx1250) — compile-verified
//
#include <hip/hip_runtime.h>
#include <stdint.h>

#define N_NODES 50000
#define N_EDGES 800000
#define D_FEAT  64
#define EDGES_PER_BLOCK 32    // 512 threads / 16 threads-per-edge
#define BLOCK_THREADS   512

// ---------------------------------------------------------------------------
// Zero-fill the output (atomics accumulate, and the harness poisons d_out).
// 50000*64 floats = 800000 float4 -> 3125 blocks * 256 threads exactly.
// ---------------------------------------------------------------------------
__global__ __launch_bounds__(256)
void zero_out_kernel(float4* __restrict__ out) {
    const int i = blockIdx.x * 256 + threadIdx.x;
    out[i] = make_float4(0.f, 0.f, 0.f, 0.f);
}

// ---------------------------------------------------------------------------
// Edge scatter: 16 threads per edge, one float4 of the 64-feature row each.
//   - edge metadata (src, dst, e) staged to LDS by wave 0 (EXEC all-ones)
//     via gfx1250 async-to-LDS + s_wait_asynccnt
//   - x row gathered with global_load_b128 (L2-resident: 35MB working set
//     vs 192MB L2)
//   - explicit non-returning global_atomic_add_f32 -> L2 atomic units
// ---------------------------------------------------------------------------
__global__ __launch_bounds__(BLOCK_THREADS)
void edge_scatter_kernel(const float* __restrict__ x,
                         const float* __restrict__ e,
                         const int*   __restrict__ src,
                         const int*   __restrict__ dst,
                         float*       __restrict__ out) {
    // LDS tile: [0..31]=src, [32..63]=dst, [64..95]=e (raw bits)
    __shared__ int tile[3 * EDGES_PER_BLOCK];

    const int tid       = threadIdx.x;
    const int edge_base = blockIdx.x * EDGES_PER_BLOCK;   // exact, no tail

    // Wave 0 (lanes 0..31, EXEC all ones): async-copy edge metadata to LDS.
    if (tid < EDGES_PER_BLOCK) {
        const uint32_t voff    = (uint32_t)(edge_base + tid) * 4u;  // byte offset
        const uint32_t lds_src = (uint32_t)(uintptr_t)&tile[tid];
        const uint32_t lds_dst = (uint32_t)(uintptr_t)&tile[EDGES_PER_BLOCK + tid];
        const uint32_t lds_e   = (uint32_t)(uintptr_t)&tile[2 * EDGES_PER_BLOCK + tid];

        asm volatile("global_load_async_to_lds_b32 %0, %1, %2"
                     :: "v"(lds_src), "v"(voff), "s"(src) : "memory");
        asm volatile("global_load_async_to_lds_b32 %0, %1, %2"
                     :: "v"(lds_dst), "v"(voff), "s"(dst) : "memory");
        asm volatile("global_load_async_to_lds_b32 %0, %1, %2"
                     :: "v"(lds_e), "v"(voff), "s"(e) : "memory");
        asm volatile("s_wait_asynccnt 0" ::: "memory");
    }
    __syncthreads();

    const int   le = tid >> 4;           // local edge 0..31
    const int   fl = (tid & 15) << 2;    // feature offset 0,4,...,60
    const int   s  = tile[le];
    const int   d  = tile[EDGES_PER_BLOCK + le];
    const float w  = __int_as_float(tile[2 * EDGES_PER_BLOCK + le]);

    // Gather one float4 of the source row (global_load_b128, hits L2).
    const float4 xv = *reinterpret_cast<const float4*>(
        x + (size_t)s * D_FEAT + fl);

    const float m0 = xv.x * w, m1 = xv.y * w, m2 = xv.z * w, m3 = xv.w * w;

    // Non-returning fp32 atomics, resolved in L2's atomic units.
    const uint32_t obyte = ((uint32_t)d * D_FEAT + (uint32_t)fl) * 4u;
    asm volatile("global_atomic_add_f32 %0, %1, %2 offset:0"
                 :: "v"(obyte), "v"(m0), "s"(out) : "memory");
    asm volatile("global_atomic_add_f32 %0, %1, %2 offset:4"
                 :: "v"(obyte), "v"(m1), "s"(out) : "memory");
    asm volatile("global_atomic_add_f32 %0, %1, %2 offset:8"
                 :: "v"(obyte), "v"(m2), "s"(out) : "memory");
    asm volatile("global_atomic_add_f32 %0, %1, %2 offset:12"
                 :: "v"(obyte), "v"(m3), "s"(out) : "memory");
}

// ---------------------------------------------------------------------------
// Inputs (setup_inputs order): t(scalar), x[N,64] f32, e[E,1] f32,
//                              src[E] i32, dst[E] i32
// Output: out[N,64] f32
// ---------------------------------------------------------------------------
extern "C" void kernel_launch(void* const* d_in, const int* in_sizes, int n_in,
                              void* d_out, int out_size, void* d_ws, size_t ws_size,
                              hipStream_t stream) {
    (void)in_sizes; (void)n_in; (void)d_ws; (void)ws_size; (void)out_size;

    const float* x   = (const float*)d_in[1];
    const float* e   = (const float*)d_in[2];
    const int*   src = (const int*)d_in[3];
    const int*   dst = (const int*)d_in[4];
    float*       out = (float*)d_out;

    // 1) zero the accumulator: 50000*64/4 = 800000 float4 -> 3125 blocks
    zero_out_kernel<<<(N_NODES * D_FEAT / 4) / 256, 256, 0, stream>>>((float4*)out);

    // 2) scatter-add messages: 800000 edges / 32 per block = 25000 blocks
    edge_scatter_kernel<<<N_EDGES / EDGES_PER_BLOCK, BLOCK_THREADS, 0, stream>>>(
        x, e, src, dst, out);
}